// Upsampling_77214922047593
// MI455X (gfx1250) — compile-verified
//
#include <hip/hip_runtime.h>
#include <hip/hip_bf16.h>

typedef float v2f __attribute__((ext_vector_type(2)));
typedef float v4f __attribute__((ext_vector_type(4)));
typedef float v8f __attribute__((ext_vector_type(8)));
typedef int   v4i __attribute__((vector_size(16)));   // matches builtin param type

#define EPSV 1e-5f
#define SLOPE 0.1f

constexpr int Bb   = 4;
constexpr int Nn   = 65536;
constexpr int Mm   = 16384;
constexpr int CIN  = 512;
constexpr int COUT = 256;
constexpr int NG   = 32;          // groups
constexpr int GS   = COUT / NG;   // 8 channels per group

constexpr int ROWS = 64;          // rows per workgroup (4 row-tiles of 16)
constexpr int RT   = ROWS / 16;   // 4 row tiles
constexpr int XLD  = CIN + 4;     // padded LDS stride for X (516)
constexpr int HLD  = COUT + 4;    // padded LDS stride for H (260)
constexpr int KP   = 32;          // K-panel depth (double buffered)
constexpr int WLD  = COUT;        // weight panel stride (row-major)
constexpr int PANEL = KP * WLD;   // 8192 floats = 32 KB

constexpr int LDS_FLOATS = ROWS * XLD + 2 * PANEL + ROWS * HLD;
constexpr size_t LDS_BYTES = (size_t)LDS_FLOATS * sizeof(float);

// ---- async-to-LDS support (guarded; falls back to sync staging) ----
#if defined(__has_builtin)
#if __has_builtin(__builtin_amdgcn_global_load_async_to_lds_b128)
#define HAVE_ASYNC 1
#endif
#endif
#ifndef HAVE_ASYNC
#define HAVE_ASYNC 0
#endif

#if HAVE_ASYNC
typedef __attribute__((address_space(1))) v4i* as1_v4i_ptr;
typedef __attribute__((address_space(3))) v4i* as3_v4i_ptr;
#endif

__device__ __forceinline__ void async_wait_all() {
#if HAVE_ASYNC
#if defined(__has_builtin) && __has_builtin(__builtin_amdgcn_s_wait_asynccnt)
    __builtin_amdgcn_s_wait_asynccnt(0);
#else
    asm volatile("s_wait_asynccnt 0x0" ::: "memory");
#endif
#endif
    __syncthreads();
}

// Copy 16 bytes global -> LDS (async if available).
__device__ __forceinline__ void copy16(const float* __restrict__ src, float* __restrict__ dst) {
#if HAVE_ASYNC
    __builtin_amdgcn_global_load_async_to_lds_b128(
        (as1_v4i_ptr)src, (as3_v4i_ptr)dst, 0, 0);
#else
    *(v4f*)dst = *(const v4f*)src;
#endif
}

// Stage one KPxCOUT weight panel: 8 x b128 per thread.
__device__ __forceinline__ void stage_panel(const float* __restrict__ src,
                                            float* __restrict__ dst, int tid) {
    #pragma unroll
    for (int i = 0; i < 8; ++i) {
        int q = (tid + i * 256) * 4;
        copy16(src + q, dst + q);
    }
}

// D = A(ROWSxK from LDS, stride lda) * W(KxCOUT global, row-major);
// result written to Hs (ROWS x HLD). Double-buffered async panel pipeline.
__device__ __forceinline__ void gemm_to_H(const float* __restrict__ Wg, int K,
                                          const float* __restrict__ As, int lda,
                                          float* __restrict__ Wbuf,   // 2*PANEL floats
                                          float* __restrict__ Hs,
                                          int tid, int lane, int wave,
                                          bool panel0_staged)
{
    v8f acc[RT][2];
    const v8f vz = {0.f,0.f,0.f,0.f,0.f,0.f,0.f,0.f};
    #pragma unroll
    for (int r = 0; r < RT; ++r) { acc[r][0] = vz; acc[r][1] = vz; }

    const int m  = lane & 15;   // row (A) / col (B) within tile
    const int kh = lane >> 4;   // K half-select

    const int nPanels = K / KP;
    if (!panel0_staged) stage_panel(Wg, Wbuf, tid);
    async_wait_all();           // panel 0 (and any prior async, e.g. X gather) ready

    for (int p = 0; p < nPanels; ++p) {
        float* cur = Wbuf + (p & 1) * PANEL;
        if (p + 1 < nPanels)    // prefetch next panel into the other buffer
            stage_panel(Wg + (size_t)(p + 1) * PANEL, Wbuf + ((p + 1) & 1) * PANEL, tid);

        const float* Ap = As + p * KP;
        #pragma unroll
        for (int k4 = 0; k4 < KP / 4; ++k4) {
            const int kk = k4 * 4 + kh * 2;      // this lane's K pair
            v2f a[RT];
            #pragma unroll
            for (int r = 0; r < RT; ++r)
                a[r] = *(const v2f*)(Ap + (r * 16 + m) * lda + kk);
            #pragma unroll
            for (int cc = 0; cc < 2; ++cc) {
                const int col = wave * 32 + cc * 16 + m;
                v2f b;
                b.x = cur[kk * WLD + col];
                b.y = cur[(kk + 1) * WLD + col];
                #pragma unroll
                for (int r = 0; r < RT; ++r)
                    acc[r][cc] = __builtin_amdgcn_wmma_f32_16x16x4_f32(
                        false, a[r], false, b, (short)0, acc[r][cc], false, false);
            }
        }
        async_wait_all();       // next panel landed; all waves done reading cur/As
    }

    // ---- spill accumulators: D layout lane->(M = 8*(lane>>4)+j, N = lane&15) ----
    const int half = lane >> 4;
    #pragma unroll
    for (int r = 0; r < RT; ++r) {
        #pragma unroll
        for (int cc = 0; cc < 2; ++cc) {
            float* dst = Hs + (r * 16 + half * 8) * HLD + wave * 32 + cc * 16 + m;
            #pragma unroll
            for (int j = 0; j < 8; ++j) dst[j * HLD] = acc[r][cc][j];
        }
    }
    __syncthreads();
}

// bias + GroupNorm(8ch) + LeakyReLU; optional query-add after activation;
// writes back to Hs, or to gout (global) for the final layer.
__device__ __forceinline__ void gn_act(float* __restrict__ Hs,
                                       const float* __restrict__ bias,
                                       const float* __restrict__ gamma,
                                       const float* __restrict__ beta,
                                       const float* __restrict__ qadd,
                                       float* __restrict__ gout,
                                       int tid, long row0)
{
    #pragma unroll
    for (int it = 0; it < (ROWS * NG) / 256; ++it) {
        int pair = tid + it * 256;
        int row  = pair >> 5;      // / NG
        int grp  = pair & (NG - 1);
        int c0   = grp * GS;
        float v[GS];
        float mu = 0.f;
        #pragma unroll
        for (int i = 0; i < GS; ++i) {
            v[i] = Hs[row * HLD + c0 + i] + bias[c0 + i];
            mu += v[i];
        }
        mu *= (1.f / GS);
        float var = 0.f;
        #pragma unroll
        for (int i = 0; i < GS; ++i) { float d = v[i] - mu; var += d * d; }
        var *= (1.f / GS);
        float rs = __frsqrt_rn(var + EPSV);
        #pragma unroll
        for (int i = 0; i < GS; ++i) {
            float h = (v[i] - mu) * rs * gamma[c0 + i] + beta[c0 + i];
            h = (h >= 0.f) ? h : SLOPE * h;
            if (qadd) h += qadd[(row0 + row) * COUT + c0 + i];
            if (gout) gout[(row0 + row) * COUT + c0 + i] = h;
            else      Hs[row * HLD + c0 + i] = h;
        }
    }
    __syncthreads();
}

__global__ __launch_bounds__(256)
void upsampling_fused_kernel(const float* __restrict__ query,
                             const float* __restrict__ support,
                             const int*   __restrict__ idx,
                             const float* __restrict__ W1, const float* __restrict__ b1,
                             const float* __restrict__ g1, const float* __restrict__ be1,
                             const float* __restrict__ W2, const float* __restrict__ b2,
                             const float* __restrict__ g2, const float* __restrict__ be2,
                             const float* __restrict__ W3, const float* __restrict__ b3,
                             const float* __restrict__ g3, const float* __restrict__ be3,
                             float* __restrict__ out)
{
    extern __shared__ float lds[];
    float* Xs   = lds;                    // ROWS x XLD (gathered latent)
    float* Wbuf = Xs + ROWS * XLD;        // 2 x (KP x WLD) weight panels
    float* Hs   = Wbuf + 2 * PANEL;       // ROWS x HLD (intermediate)

    const int tid  = threadIdx.x;
    const int lane = tid & 31;
    const int wave = tid >> 5;
    const long row0 = (long)blockIdx.x * ROWS;   // global row in [0, B*N)

    // ---- gather latent rows: 4 threads per row, 128 contiguous floats each ----
    {
        int r = tid >> 2;
        int cbase = (tid & 3) * 128;
        long g = row0 + r;
        int b = (int)(g / Nn);
        int srow = idx[g];
        const float* src = support + ((size_t)b * Mm + srow) * CIN + cbase;
        float* dst = Xs + r * XLD + cbase;
        #pragma unroll
        for (int i = 0; i < 128; i += 4)
            copy16(src + i, dst + i);
    }
    // Issue W1 panel 0 alongside the gather; one wait covers both.
    stage_panel(W1, Wbuf, tid);

    // Layer 1: (64x512) x (512x256)
    gemm_to_H(W1, CIN, Xs, XLD, Wbuf, Hs, tid, lane, wave, /*panel0_staged=*/true);
    gn_act(Hs, b1, g1, be1, nullptr, nullptr, tid, row0);

    // Layer 2: (64x256) x (256x256), then += query after activation
    gemm_to_H(W2, COUT, Hs, HLD, Wbuf, Hs, tid, lane, wave, false);
    gn_act(Hs, b2, g2, be2, query, nullptr, tid, row0);

    // Layer 3: (64x256) x (256x256), final GN+act streamed to global out
    gemm_to_H(W3, COUT, Hs, HLD, Wbuf, Hs, tid, lane, wave, false);
    gn_act(Hs, b3, g3, be3, nullptr, out, tid, row0);
}

extern "C" void kernel_launch(void* const* d_in, const int* in_sizes, int n_in,
                              void* d_out, int out_size, void* d_ws, size_t ws_size,
                              hipStream_t stream) {
    const float* query   = (const float*)d_in[0];
    const float* support = (const float*)d_in[1];
    const int*   idx     = (const int*)  d_in[2];
    const float* W1  = (const float*)d_in[3];
    const float* b1  = (const float*)d_in[4];
    const float* g1  = (const float*)d_in[5];
    const float* be1 = (const float*)d_in[6];
    const float* W2  = (const float*)d_in[7];
    const float* b2  = (const float*)d_in[8];
    const float* g2  = (const float*)d_in[9];
    const float* be2 = (const float*)d_in[10];
    const float* W3  = (const float*)d_in[11];
    const float* b3  = (const float*)d_in[12];
    const float* g3  = (const float*)d_in[13];
    const float* be3 = (const float*)d_in[14];
    float* out = (float*)d_out;

    (void)in_sizes; (void)n_in; (void)out_size; (void)d_ws; (void)ws_size;

    (void)hipFuncSetAttribute((const void*)upsampling_fused_kernel,
                              hipFuncAttributeMaxDynamicSharedMemorySize,
                              (int)LDS_BYTES);

    dim3 grid((Bb * Nn) / ROWS);   // 4096 blocks
    dim3 block(256);               // 8 wave32 waves
    upsampling_fused_kernel<<<grid, block, LDS_BYTES, stream>>>(
        query, support, idx,
        W1, b1, g1, be1,
        W2, b2, g2, be2,
        W3, b3, g3, be3,
        out);
}